// HashEmbeddingEncoder_11587821765177
// MI455X (gfx1250) — compile-verified
//
#include <hip/hip_runtime.h>

// ---------------------------------------------------------------------------
// Multi-resolution hash-grid encoder with sign binarization (forward value).
//   out[n, l*8 + f] = sign(tables[l, hash(cell(n,l)) & (2^19-1), f])
// Memory-bound: ~588 MB HBM traffic -> ~25us floor @ 23.3 TB/s.
// Key MI455X-specific choice: NT stores for the 384MB output stream so the
// 192MB table footprint (== global L2 size) stays resident in L2.
// ---------------------------------------------------------------------------

#define HE_NUM_LEVELS 12
#define HE_FEAT 8
#define HE_TABLE_SIZE (1u << 19)
#define HE_TABLE_MASK (HE_TABLE_SIZE - 1u)

typedef float v4f __attribute__((ext_vector_type(4)));

// resolutions = int(16 * 32**(l/11)), computed in float64 by the reference
__constant__ int c_res[HE_NUM_LEVELS] = {16, 21, 30, 41, 56, 77, 105, 145, 198, 272, 373, 512};

__device__ __forceinline__ float sgn(float x) {
    // jnp.sign semantics for finite inputs: +1 / -1 / 0
    return (x > 0.0f) ? 1.0f : ((x < 0.0f) ? -1.0f : 0.0f);
}

__global__ __launch_bounds__(256) void hash_encode_sign_kernel(
    const float* __restrict__ coords,   // [N,3]
    const float* __restrict__ tables,   // [12, 2^19, 8]
    float* __restrict__ out,            // [N, 96]
    int npairs)                         // N * 12
{
    int pid = blockIdx.x * 256 + threadIdx.x;
    if (pid >= npairs) return;

    int p = pid / HE_NUM_LEVELS;        // point index (const-div -> mulhi)
    int l = pid - p * HE_NUM_LEVELS;    // level index

    int   resi = c_res[l];
    float resf = (float)resi;
    int   rmax = resi - 1;

    // coords for this point; 12 lanes share a point -> L0-cached broadcasts
    float cx = coords[3 * p + 0];
    float cy = coords[3 * p + 1];
    float cz = coords[3 * p + 2];

    // cell index: clamp(floor(c*res), 0, res-1)  (f32 mul/floor matches ref)
    int ix = (int)floorf(cx * resf);
    int iy = (int)floorf(cy * resf);
    int iz = (int)floorf(cz * resf);
    ix = ix < 0 ? 0 : (ix > rmax ? rmax : ix);
    iy = iy < 0 ? 0 : (iy > rmax ? rmax : iy);
    iz = iz < 0 ? 0 : (iz > rmax ? rmax : iz);

    // spatial hash: (x*1 ^ y*2654435761 ^ z*805459861) mod 2^19.
    // The int64 math in the reference only affects bits >= 19 which are
    // masked away, so 32-bit multiplies are bit-exact here.
    unsigned h = (unsigned)ix
               ^ ((unsigned)iy * 2654435761u)
               ^ ((unsigned)iz * 805459861u);
    unsigned idx = h & HE_TABLE_MASK;

    // gather 8 floats (32B) from this level's table; rows are 32B so 16B
    // alignment holds -> two global_load_b128. Default RT hint: keep the
    // (L2-sized) tables resident in L2.
    const v4f* row = (const v4f*)(tables +
        ((size_t)l * HE_TABLE_SIZE + idx) * HE_FEAT);
    v4f a = row[0];
    v4f b = row[1];

    v4f sa, sb;
    sa.x = sgn(a.x); sa.y = sgn(a.y); sa.z = sgn(a.z); sa.w = sgn(a.w);
    sb.x = sgn(b.x); sb.y = sgn(b.y); sb.z = sgn(b.z); sb.w = sgn(b.w);

    // Output offset: n*96 + l*8 == pid*8 -> globally contiguous 32B/thread.
    // Non-temporal stores (th:TH_STORE_NT): the 384MB output is write-once,
    // never re-read; don't let it evict the table working set from L2.
    v4f* o = (v4f*)(out + (size_t)pid * HE_FEAT);
    __builtin_nontemporal_store(sa, o);
    __builtin_nontemporal_store(sb, o + 1);
}

extern "C" void kernel_launch(void* const* d_in, const int* in_sizes, int n_in,
                              void* d_out, int out_size, void* d_ws, size_t ws_size,
                              hipStream_t stream) {
    const float* coords = (const float*)d_in[0];   // [N,3] f32
    const float* tables = (const float*)d_in[1];   // [12, 2^19, 8] f32
    float* out = (float*)d_out;                    // [N, 96] f32

    int npoints = in_sizes[0] / 3;
    int npairs  = npoints * HE_NUM_LEVELS;

    int threads = 256;
    int blocks  = (npairs + threads - 1) / threads;
    hash_encode_sign_kernel<<<blocks, threads, 0, stream>>>(coords, tables, out, npairs);
}